// SPRSConv3dTestTorch_77567109366052
// MI455X (gfx1250) — compile-verified
//
#include <hip/hip_runtime.h>
#include <hip/hip_bf16.h>

// ---------------- problem constants (from reference) ----------------
#define N_PTS   400000
#define CIN     32
#define COUT    64
#define OSZ     100
#define OSY     100
#define OSX     8
#define NUM_OUT (2 * OSZ * OSY * OSX)   // 160000
#define RANK    200000                  // int(N * 0.5)
#define PACKB_BYTES (27 * 4 * 8 * 32 * 8)   // 221184
#define MAX_SLOTS (N_PTS + 8 * 15)          // 400120 padded slots
#define N_TILES 25008                       // ceil(MAX_SLOTS/16), 3126 blocks * 8 waves

typedef __attribute__((ext_vector_type(2))) float v2f;
typedef __attribute__((ext_vector_type(8))) float v8f;

// ---------------- helpers ----------------
__device__ __forceinline__ unsigned fkey(float f) {
  unsigned u = __float_as_uint(f);
  return (u & 0x80000000u) ? ~u : (u | 0x80000000u);
}

// ---------------- zero workspace / output ----------------
__global__ void zero_kernel(float4* __restrict__ acc, unsigned* __restrict__ imp,
                            unsigned* __restrict__ hist, unsigned* __restrict__ hist2,
                            unsigned* __restrict__ misc) {
  int i = blockIdx.x * blockDim.x + threadIdx.x;
  int stride = gridDim.x * blockDim.x;
  for (int j = i; j < NUM_OUT * COUT / 4; j += stride)
    acc[j] = make_float4(0.f, 0.f, 0.f, 0.f);
  for (int j = i; j < NUM_OUT; j += stride) imp[j] = 0u;
  for (int j = i; j < 65536; j += stride) { hist[j] = 0u; hist2[j] = 0u; }
  if (i < 64) misc[i] = 0u;   // sel/counts/fill/meta scratch
}

// ---------------- repack weights into WMMA B-fragment order ----------------
// packB[((off*4 + nt)*8 + k)*32 + lane] = { W[off][4k+2h][nt*16+n], W[off][4k+2h+1][nt*16+n] }
// with lane = h*16 + n  (V_WMMA_F32_16X16X4_F32 B layout: v0 = K0|K2, v1 = K1|K3)
__global__ void repack_kernel(const float* __restrict__ W, float2* __restrict__ packB) {
  int idx = blockIdx.x * blockDim.x + threadIdx.x;
  if (idx >= 27 * 4 * 8 * 32) return;
  int lane = idx & 31;
  int k    = (idx >> 5) & 7;
  int nt   = (idx >> 8) & 3;
  int off  = idx >> 10;
  int h = lane >> 4, n = lane & 15;
  int row = 4 * k + 2 * h;
  int col = nt * 16 + n;
  const float* wb = W + (size_t)off * CIN * COUT;
  packB[idx] = make_float2(wb[row * COUT + col], wb[(row + 1) * COUT + col]);
}

// ---------------- exact median: 2-pass 16-bit radix select ----------------
__global__ void hist_hi_kernel(const float* __restrict__ mask, unsigned* __restrict__ hist) {
  int i = blockIdx.x * blockDim.x + threadIdx.x;
  int stride = gridDim.x * blockDim.x;
  for (int j = i; j < N_PTS; j += stride)
    atomicAdd(&hist[fkey(mask[j]) >> 16], 1u);
}

__global__ void hist_lo_kernel(const float* __restrict__ mask,
                               const unsigned* __restrict__ sel,
                               unsigned* __restrict__ hist2) {
  unsigned hb = sel[0];
  int i = blockIdx.x * blockDim.x + threadIdx.x;
  int stride = gridDim.x * blockDim.x;
  for (int j = i; j < N_PTS; j += stride) {
    unsigned k = fkey(mask[j]);
    if ((k >> 16) == hb) atomicAdd(&hist2[k & 0xFFFFu], 1u);
  }
}

// single block, 256 threads; which==0 -> hi bin, which==1 -> lo bin + threshold bits
__global__ void select_kernel(const unsigned* __restrict__ hist, unsigned* __restrict__ sel,
                              int which) {
  __shared__ unsigned s[256];
  int t = threadIdx.x;
  unsigned sum = 0;
  for (int j = 0; j < 256; j++) sum += hist[t * 256 + j];
  s[t] = sum;
  __syncthreads();
  if (t == 0) {
    unsigned rank = (which == 0) ? (unsigned)RANK : sel[1];
    unsigned cum = 0;
    int c = 0;
    for (; c < 256; c++) { if (cum + s[c] > rank) break; cum += s[c]; }
    unsigned rank2 = rank - cum;
    unsigned cum2 = 0;
    int b = 0;
    for (; b < 256; b++) { unsigned hc = hist[c * 256 + b]; if (cum2 + hc > rank2) break; cum2 += hc; }
    unsigned bin = (unsigned)c * 256u + (unsigned)b;
    if (which == 0) {
      sel[0] = bin;
      sel[1] = rank2 - cum2;
    } else {
      unsigned key = (sel[0] << 16) | bin;
      unsigned u = (key & 0x80000000u) ? (key ^ 0x80000000u) : ~key;
      sel[2] = u;
    }
  }
}

// ---------------- parity binning: count -> aligned prefix -> scatter ----------------
__device__ __forceinline__ int parity_key(int pz, int py, int px) {
  return (((pz + 1) & 1) << 2) | (((py + 1) & 1) << 1) | ((px + 1) & 1);
}

__global__ void parity_count_kernel(const int* __restrict__ coors, unsigned* __restrict__ counts) {
  int i = blockIdx.x * blockDim.x + threadIdx.x;
  int stride = gridDim.x * blockDim.x;
  for (int j = i; j < N_PTS; j += stride) {
    int4 c = *(const int4*)(coors + (size_t)j * 4);
    atomicAdd(&counts[parity_key(c.y, c.z, c.w)], 1u);
  }
}

// meta[i] = 16-aligned start[i] (i=0..8), meta[16+i] = end[i] = start[i]+count[i]
__global__ void parity_prefix_kernel(const unsigned* __restrict__ counts,
                                     unsigned* __restrict__ meta) {
  if (threadIdx.x == 0 && blockIdx.x == 0) {
    unsigned s = 0;
    for (int i = 0; i < 8; i++) {
      meta[i] = s;
      meta[16 + i] = s + counts[i];
      s += (counts[i] + 15u) & ~15u;
    }
    meta[8] = s;
  }
}

__global__ void parity_scatter_kernel(const int* __restrict__ coors,
                                      const unsigned* __restrict__ meta,
                                      unsigned* __restrict__ fill,
                                      unsigned* __restrict__ perm) {
  int i = blockIdx.x * blockDim.x + threadIdx.x;
  int stride = gridDim.x * blockDim.x;
  for (int j = i; j < N_PTS; j += stride) {
    int4 c = *(const int4*)(coors + (size_t)j * 4);
    int key = parity_key(c.y, c.z, c.w);
    unsigned pos = meta[key] + atomicAdd(&fill[key], 1u);
    perm[pos] = (unsigned)j;
  }
}

// ---------------- main sparse-conv kernel: WMMA f32 16x16x4, LDS-staged B ----------------
__global__ __launch_bounds__(256)
void sprs_conv_kernel(const float*  __restrict__ feat,
                      const int*    __restrict__ coors,
                      const float*  __restrict__ mask,
                      const float2* __restrict__ packB,
                      const unsigned* __restrict__ sel,
                      const unsigned* __restrict__ meta,
                      const unsigned* __restrict__ perm,
                      float*        __restrict__ acc,
                      unsigned*     __restrict__ imp) {
  extern __shared__ char smem[];          // 221184 B: whole packB, staged once per block
  const int tid = threadIdx.x;

  // async global -> LDS copy of the repacked weights (ASYNCcnt path)
  {
    const char* gsrc = (const char*)packB;
    for (int j = tid; j < PACKB_BYTES / 16; j += 256) {
      unsigned ldsoff = (unsigned)(size_t)&smem[j * 16];
      unsigned long long gaddr = (unsigned long long)(size_t)(gsrc + j * 16);
      asm volatile("global_load_async_to_lds_b128 %0, %1, off"
                   :: "v"(ldsoff), "v"(gaddr) : "memory");
    }
    asm volatile("s_wait_asynccnt 0x0" ::: "memory");
    __syncthreads();
  }

  const int lane = tid & 31;
  const int wave = tid >> 5;
  const int tile = blockIdx.x * 8 + wave;     // 25008 tiles
  const int m = lane & 15;
  const int h = lane >> 4;
  const int slot = tile * 16 + m;

  // bin lookup: starts are 16-aligned, so a tile lies entirely in one bin (wave-uniform)
  int bin = 0;
#pragma unroll
  for (int i = 1; i < 8; i++)
    if (slot >= (int)meta[i]) bin = i;
  const int rowReal = slot < (int)meta[16 + bin];
  const int pt = rowReal ? (int)perm[slot] : 0;

  // A fragments: 16x32 fp32, per the 16x4 f32 WMMA layout
  v2f a[8];
  const float* fp = feat + (size_t)pt * CIN + 2 * h;
#pragma unroll
  for (int k = 0; k < 8; k++) a[k] = *(const v2f*)(fp + 4 * k);

  const int4 cz = *(const int4*)(coors + (size_t)pt * 4);
  const int bcoord = cz.x, pz = cz.y, py = cz.z, px = cz.w;
  const float thr = __uint_as_float(sel[2]);
  const int important = (mask[pt] >= thr) ? 1 : 0;

  // this tile's parity -> only the offsets that pass the stride-2 parity test
  const int bz = (bin >> 2) & 1, by = (bin >> 1) & 1, bx = bin & 1;
  const int n0 = bz ? 1 : 2, n1 = by ? 1 : 2, n2 = bx ? 1 : 2;

  for (int i0 = 0; i0 < n0; i0++) {
    const int o0 = (bz ? 1 : 0) + 2 * i0;
    for (int i1 = 0; i1 < n1; i1++) {
      const int o1 = (by ? 1 : 0) + 2 * i1;
      for (int i2 = 0; i2 < n2; i2++) {
        const int o2 = (bx ? 1 : 0) + 2 * i2;
        const int off = (o0 * 3 + o1) * 3 + o2;

        const int ocz = (pz + 1 - o0) >> 1;
        const int ocy = (py + 1 - o1) >> 1;
        const int ocx = (px + 1 - o2) >> 1;
        const bool valid = rowReal &&
                           (ocz >= 0) && (ocz < OSZ) &&
                           (ocy >= 0) && (ocy < OSY) &&
                           (ocx >= 0) && (ocx < OSX);
        const int lin = ((bcoord * OSZ + ocz) * OSY + ocy) * OSX + ocx;

        const unsigned long long bal = __ballot(valid);
        if (bal == 0ull) continue;               // wave-uniform skip

        int vr[8], lr[8];
#pragma unroll
        for (int j = 0; j < 8; j++) {
          const int r = j + 8 * h;
          vr[j] = __shfl(valid ? 1 : 0, r, 32);
          lr[j] = __shfl(lin, r, 32);
        }

        const float2* bp = (const float2*)smem + (size_t)off * (4 * 8 * 32);
#pragma unroll
        for (int nt = 0; nt < 4; ++nt) {
          v8f c = {};
          const float2* bpt = bp + nt * (8 * 32) + lane;
#pragma unroll
          for (int k = 0; k < 8; k++) {
            const v2f b = *(const v2f*)(bpt + k * 32);
            c = __builtin_amdgcn_wmma_f32_16x16x4_f32(
                    false, a[k], false, b, (short)0, c, false, false);
          }
#pragma unroll
          for (int j = 0; j < 8; j++) {
            if (vr[j]) {
              unsafeAtomicAdd(acc + (size_t)lr[j] * COUT + nt * 16 + m, c[j]);
            }
          }
        }

        if (h == 0 && valid && important) atomicAdd(imp + lin, 1u);
      }
    }
  }
}

// ---------------- finalize: out *= (imp_cnt > 0) ----------------
__global__ void finalize_kernel(float4* __restrict__ out, const unsigned* __restrict__ imp) {
  int i = blockIdx.x * blockDim.x + threadIdx.x;   // NUM_OUT*COUT/4 threads
  float4 v = out[i];
  float s = (imp[i >> 4] != 0u) ? 1.0f : 0.0f;     // 16 float4 per voxel
  v.x *= s; v.y *= s; v.z *= s; v.w *= s;
  out[i] = v;
}

// ---------------- launch ----------------
extern "C" void kernel_launch(void* const* d_in, const int* in_sizes, int n_in,
                              void* d_out, int out_size, void* d_ws, size_t ws_size,
                              hipStream_t stream) {
  const float* feat   = (const float*)d_in[0];
  const int*   coors  = (const int*)d_in[1];
  const float* mask   = (const float*)d_in[2];
  const float* weight = (const float*)d_in[3];
  (void)in_sizes; (void)n_in; (void)out_size; (void)ws_size;

  // workspace layout; d_out doubles as the accumulator (dump slot never written)
  char* ws = (char*)d_ws;
  unsigned* imp    = (unsigned*)(ws);                         // 640000 B (reserve 640K)
  unsigned* hist   = (unsigned*)(ws + 640 * 1024);            // 256 KiB
  unsigned* hist2  = (unsigned*)(ws + 896 * 1024);            // 256 KiB
  unsigned* misc   = (unsigned*)(ws + 1152 * 1024);           // 64 u32: sel/counts/fill/meta
  unsigned* sel    = misc;            // [0..2]
  unsigned* counts = misc + 4;        // [4..11]
  unsigned* fill   = misc + 12;       // [12..19]
  unsigned* meta   = misc + 20;       // start[0..8] at meta[0..8], end[i] at meta[16+i]
  float2*   packB  = (float2*)(ws + 1153 * 1024);             // 221184 B (reserve 224K)
  unsigned* perm   = (unsigned*)(ws + 1377 * 1024);           // 400120 u32 (~1563K)
  float*    acc    = (float*)d_out;

  zero_kernel<<<4096, 256, 0, stream>>>((float4*)acc, imp, hist, hist2, misc);
  repack_kernel<<<108, 256, 0, stream>>>(weight, packB);
  hist_hi_kernel<<<1024, 256, 0, stream>>>(mask, hist);
  select_kernel<<<1, 256, 0, stream>>>(hist, sel, 0);
  hist_lo_kernel<<<1024, 256, 0, stream>>>(mask, sel, hist2);
  select_kernel<<<1, 256, 0, stream>>>(hist2, sel, 1);
  parity_count_kernel<<<1024, 256, 0, stream>>>(coors, counts);
  parity_prefix_kernel<<<1, 64, 0, stream>>>(counts, meta);
  parity_scatter_kernel<<<1024, 256, 0, stream>>>(coors, meta, fill, perm);
  sprs_conv_kernel<<<N_TILES / 8, 256, PACKB_BYTES, stream>>>(
      feat, coors, mask, packB, sel, meta, perm, acc, imp);
  finalize_kernel<<<NUM_OUT * COUT / 4 / 256, 256, 0, stream>>>((float4*)acc, imp);
}